// emb_att_Layers_53609781788702
// MI455X (gfx1250) — compile-verified
//
#include <hip/hip_runtime.h>
#include <math.h>

typedef float v2f __attribute__((ext_vector_type(2)));
typedef float v8f __attribute__((ext_vector_type(8)));

#define D_FEAT 64
#define NRELS  16
#define N_SEQ  4
#define DHEAD  16
#define KTOT   (NRELS * D_FEAT + D_FEAT)   // 1088

// ------------------------------------------------------------------ fill ----
// count must be a multiple of 4 (all our regions are).
__global__ void fill_zero_kernel(float4* __restrict__ p, long long count4) {
  long long i = (long long)blockIdx.x * blockDim.x + threadIdx.x;
  long long stride = (long long)gridDim.x * blockDim.x;
  float4 z; z.x = z.y = z.z = z.w = 0.0f;
  for (; i < count4; i += stride) p[i] = z;
}

// ------------------------------------------------------- B-operand pack ----
// bpack[(k>>1)*COUT*2 + col*2 + (k&1)] = wcat[k][col], wcat = [w ; root]
__global__ void pack_b_kernel(const float* __restrict__ w,
                              const float* __restrict__ root,
                              float* __restrict__ bpack, int cout) {
  int idx = blockIdx.x * blockDim.x + threadIdx.x;
  int total = KTOT * cout;
  if (idx >= total) return;
  int k = idx / cout;
  int col = idx - k * cout;
  float v = (k < NRELS * D_FEAT) ? w[(size_t)k * cout + col]
                                 : root[(size_t)(k - NRELS * D_FEAT) * cout + col];
  bpack[((size_t)(k >> 1) * cout + col) * 2 + (k & 1)] = v;
}

// ------------------------------------------------------------------- MHA ----
// 64 threads / block, 4 nodes / block. Thread j owns channel j.
__global__ __launch_bounds__(64) void mha_kernel(
    const float* __restrict__ emb, const float* __restrict__ in_w,
    const float* __restrict__ in_b, const float* __restrict__ out_w,
    const float* __restrict__ out_b, float* __restrict__ x0, int N) {
  __shared__ float xs[N_SEQ][4][D_FEAT];   // [s][t][d]
  __shared__ float qb[4][D_FEAT];
  __shared__ float kb[4][N_SEQ][D_FEAT];
  __shared__ float vb[4][N_SEQ][D_FEAT];
  __shared__ float sc[4][16];              // [t][h*4+s]
  __shared__ float ob[4][D_FEAT];

  const int j  = threadIdx.x;
  const int n0 = blockIdx.x * 4;

  for (int idx = j; idx < N_SEQ * 4 * D_FEAT; idx += 64) {
    int s  = idx / (4 * D_FEAT);
    int t  = (idx / D_FEAT) & 3;
    int dd = idx & (D_FEAT - 1);
    int node = n0 + t;
    xs[s][t][dd] = (node < N) ? emb[((size_t)s * N + node) * D_FEAT + dd] : 0.0f;
  }
  __syncthreads();

  float qa[4] = {0.f, 0.f, 0.f, 0.f};
  float ka[4][N_SEQ] = {};
  float va[4][N_SEQ] = {};
  const float* Wq = in_w + (size_t)j * D_FEAT;
  const float* Wk = in_w + (size_t)(D_FEAT + j) * D_FEAT;
  const float* Wv = in_w + (size_t)(2 * D_FEAT + j) * D_FEAT;
  for (int d = 0; d < D_FEAT; ++d) {
    float wq = Wq[d], wk = Wk[d], wv = Wv[d];
    #pragma unroll
    for (int t = 0; t < 4; ++t) {
      qa[t] = fmaf(wq, xs[0][t][d], qa[t]);
      #pragma unroll
      for (int s = 0; s < N_SEQ; ++s) {
        float xv = xs[s][t][d];
        ka[t][s] = fmaf(wk, xv, ka[t][s]);
        va[t][s] = fmaf(wv, xv, va[t][s]);
      }
    }
  }
  {
    float bq = in_b[j], bk = in_b[D_FEAT + j], bv = in_b[2 * D_FEAT + j];
    #pragma unroll
    for (int t = 0; t < 4; ++t) {
      qb[t][j] = qa[t] + bq;
      #pragma unroll
      for (int s = 0; s < N_SEQ; ++s) {
        kb[t][s][j] = ka[t][s] + bk;
        vb[t][s][j] = va[t][s] + bv;
      }
    }
  }
  __syncthreads();

  {   // scores: thread j -> (t, h, s)
    int t = j >> 4, h = (j >> 2) & 3, s = j & 3;
    float acc = 0.f;
    #pragma unroll
    for (int i = 0; i < DHEAD; ++i)
      acc = fmaf(qb[t][h * DHEAD + i], kb[t][s][h * DHEAD + i], acc);
    sc[t][h * 4 + s] = acc * 0.25f;   // 1/sqrt(16)
  }
  __syncthreads();
  if (j < 16) {                       // softmax per (t, h)
    int t = j >> 2, h = j & 3;
    float m = -1e30f;
    #pragma unroll
    for (int s = 0; s < N_SEQ; ++s) m = fmaxf(m, sc[t][h * 4 + s]);
    float e[N_SEQ], sum = 0.f;
    #pragma unroll
    for (int s = 0; s < N_SEQ; ++s) { e[s] = __expf(sc[t][h * 4 + s] - m); sum += e[s]; }
    float inv = 1.0f / sum;
    #pragma unroll
    for (int s = 0; s < N_SEQ; ++s) sc[t][h * 4 + s] = e[s] * inv;
  }
  __syncthreads();
  {                                   // o = attn @ v
    int h = j >> 4;
    #pragma unroll
    for (int t = 0; t < 4; ++t) {
      float acc = 0.f;
      #pragma unroll
      for (int s = 0; s < N_SEQ; ++s) acc = fmaf(sc[t][h * 4 + s], vb[t][s][j], acc);
      ob[t][j] = acc;
    }
  }
  __syncthreads();
  {                                   // out proj
    float acc[4] = {0.f, 0.f, 0.f, 0.f};
    const float* Wr = out_w + (size_t)j * D_FEAT;
    for (int i = 0; i < D_FEAT; ++i) {
      float w = Wr[i];
      #pragma unroll
      for (int t = 0; t < 4; ++t) acc[t] = fmaf(w, ob[t][i], acc[t]);
    }
    float bo = out_b[j];
    #pragma unroll
    for (int t = 0; t < 4; ++t) {
      int node = n0 + t;
      if (node < N) x0[(size_t)node * D_FEAT + j] = acc[t] + bo;
    }
  }
}

// ----------------------------------------------------------- aggregation ----
// 256 threads = 4 edges x 64 channels. Native f32 atomics into (dst,rel) segs.
__global__ __launch_bounds__(256) void edge_agg_kernel(
    const float* __restrict__ x, const int* __restrict__ src,
    const int* __restrict__ dst, const int* __restrict__ et,
    float* __restrict__ agg, float* __restrict__ cnt, int E, int addCnt) {
  int e = blockIdx.x * 4 + (threadIdx.x >> 6);
  int c = threadIdx.x & 63;
  if (e >= E) return;
  int s = src[e];
  int d = dst[e];
  int t = et[e];
  float val = x[(size_t)s * 64 + c];
  float* addr = agg + ((size_t)d * NRELS + t) * 64 + c;
  __hip_atomic_fetch_add(addr, val, __ATOMIC_RELAXED, __HIP_MEMORY_SCOPE_AGENT);
  if (addCnt && c == 0)
    __hip_atomic_fetch_add(cnt + (size_t)d * NRELS + t, 1.0f,
                           __ATOMIC_RELAXED, __HIP_MEMORY_SCOPE_AGENT);
}

// -------------------------------------------------- WMMA einsum (+ root) ----
// out[n,o] = sum_{r,i} (agg[n,r,i]/max(cnt,1)) w[r,i,o] + x[n]@root + b
// 4 waves/block, each wave a 32-node x COUT tile, K = 1088 via f32 WMMA.
__device__ __forceinline__ float sigmoid_f(float v) {
  return 1.0f / (1.0f + __expf(-v));
}

template <int COUT, int ACT>   // ACT: 0 = relu, 1 = sigmoid
__global__ __launch_bounds__(128) void rgcn_einsum_kernel(
    const float* __restrict__ agg, const float* __restrict__ cnt,
    const float* __restrict__ x, const float* __restrict__ bpack,
    const float* __restrict__ bias, float* __restrict__ out, int N) {
  constexpr int OT = COUT / 16;
  const int lane  = threadIdx.x & 31;
  const int wave  = threadIdx.x >> 5;
  const int n0    = (blockIdx.x * 4 + wave) * 32;
  if (n0 >= N) return;                      // wave-uniform
  const int mrow  = lane & 15;
  const int khalf = (lane >> 4) << 1;       // 0 or 2

  const v8f vzero = {0.f, 0.f, 0.f, 0.f, 0.f, 0.f, 0.f, 0.f};
  v8f acc[2][OT];
  #pragma unroll
  for (int a = 0; a < 2; ++a)
    #pragma unroll
    for (int b = 0; b < OT; ++b) acc[a][b] = vzero;

  const int nrow0 = n0 + mrow;
  const int nrow1 = n0 + 16 + mrow;
  const int ncl0 = nrow0 < N ? nrow0 : N - 1;   // clamp: loads always valid
  const int ncl1 = nrow1 < N ? nrow1 : N - 1;

  const float* pa0 = agg + (size_t)ncl0 * (NRELS * D_FEAT) + khalf;
  const float* pa1 = agg + (size_t)ncl1 * (NRELS * D_FEAT) + khalf;
  const float* pc0 = cnt + (size_t)ncl0 * NRELS;
  const float* pc1 = cnt + (size_t)ncl1 * NRELS;
  const float* pb  = bpack + (size_t)(khalf >> 1) * (COUT * 2);

  // ---- relation region: k in [0, 1024) ----
  for (int r = 0; r < NRELS; ++r) {
    const float invc0 = 1.0f / fmaxf(pc0[r], 1.0f);
    const float invc1 = 1.0f / fmaxf(pc1[r], 1.0f);
    const float* qa0 = pa0 + r * D_FEAT;
    const float* qa1 = pa1 + r * D_FEAT;
    const float* qb  = pb + (size_t)r * 32 * (COUT * 2);
    __builtin_prefetch(qa0 + D_FEAT, 0, 1);   // next relation block
    __builtin_prefetch(qa1 + D_FEAT, 0, 1);
    #pragma unroll
    for (int step = 0; step < 16; ++step) {
      v2f a0 = *(const v2f*)(qa0 + step * 4);
      v2f a1 = *(const v2f*)(qa1 + step * 4);
      a0.x *= invc0; a0.y *= invc0;
      a1.x *= invc1; a1.y *= invc1;
      #pragma unroll
      for (int ot = 0; ot < OT; ++ot) {
        int col = ot * 16 + mrow;
        v2f b = *(const v2f*)(qb + (size_t)step * 2 * (COUT * 2) + col * 2);
        acc[0][ot] = __builtin_amdgcn_wmma_f32_16x16x4_f32(
            false, a0, false, b, (short)0, acc[0][ot], false, false);
        acc[1][ot] = __builtin_amdgcn_wmma_f32_16x16x4_f32(
            false, a1, false, b, (short)0, acc[1][ot], false, false);
      }
    }
  }

  // ---- root region: k in [1024, 1088), A = x ----
  {
    const float* qa0 = x + (size_t)ncl0 * D_FEAT + khalf;
    const float* qa1 = x + (size_t)ncl1 * D_FEAT + khalf;
    const float* qb  = pb + (size_t)(NRELS * 32) * (COUT * 2);
    #pragma unroll
    for (int step = 0; step < 16; ++step) {
      v2f a0 = *(const v2f*)(qa0 + step * 4);
      v2f a1 = *(const v2f*)(qa1 + step * 4);
      #pragma unroll
      for (int ot = 0; ot < OT; ++ot) {
        int col = ot * 16 + mrow;
        v2f b = *(const v2f*)(qb + (size_t)step * 2 * (COUT * 2) + col * 2);
        acc[0][ot] = __builtin_amdgcn_wmma_f32_16x16x4_f32(
            false, a0, false, b, (short)0, acc[0][ot], false, false);
        acc[1][ot] = __builtin_amdgcn_wmma_f32_16x16x4_f32(
            false, a1, false, b, (short)0, acc[1][ot], false, false);
      }
    }
  }

  // D layout: VGPR v holds M = v + 8*(lane>=16), N-col = lane&15
  const int mbase = (lane >> 4) * 8;
  #pragma unroll
  for (int nt = 0; nt < 2; ++nt) {
    #pragma unroll
    for (int ot = 0; ot < OT; ++ot) {
      int col = ot * 16 + mrow;
      float bv = bias[col];
      #pragma unroll
      for (int vv = 0; vv < 8; ++vv) {
        int node = n0 + nt * 16 + mbase + vv;
        if (node < N) {
          float val = acc[nt][ot][vv] + bv;
          val = (ACT == 0) ? fmaxf(val, 0.0f) : sigmoid_f(val);
          out[(size_t)node * COUT + col] = val;
        }
      }
    }
  }
}

// ---------------------------------------------------------------- launch ----
extern "C" void kernel_launch(void* const* d_in, const int* in_sizes, int n_in,
                              void* d_out, int out_size, void* d_ws, size_t ws_size,
                              hipStream_t stream) {
  const float* emb   = (const float*)d_in[0];
  const int*   eidx  = (const int*)  d_in[1];
  const int*   etype = (const int*)  d_in[2];
  const float* in_w  = (const float*)d_in[3];
  const float* in_b  = (const float*)d_in[4];
  const float* out_w = (const float*)d_in[5];
  const float* out_b = (const float*)d_in[6];
  const float* w1    = (const float*)d_in[7];
  const float* root1 = (const float*)d_in[8];
  const float* b1    = (const float*)d_in[9];
  const float* w2    = (const float*)d_in[10];
  const float* root2 = (const float*)d_in[11];
  const float* b2    = (const float*)d_in[12];

  const int N = in_sizes[0] / (N_SEQ * D_FEAT);
  const int E = in_sizes[1] / 2;
  const int* src = eidx;
  const int* dst = eidx + E;

  // workspace: x0 | x1 | cnt | agg | bpack1 | bpack2
  float* x0     = (float*)d_ws;
  float* x1     = x0 + (size_t)N * D_FEAT;
  float* cnt    = x1 + (size_t)N * D_FEAT;
  float* agg    = cnt + (size_t)N * NRELS;
  float* bpack1 = agg + (size_t)N * NRELS * D_FEAT;
  float* bpack2 = bpack1 + (size_t)KTOT * 64;

  // 0) pack B operands (deterministic, cheap)
  pack_b_kernel<<<(KTOT * 64 + 255) / 256, 256, 0, stream>>>(w1, root1, bpack1, 64);
  pack_b_kernel<<<(KTOT * 16 + 255) / 256, 256, 0, stream>>>(w2, root2, bpack2, 16);

  // 1) MHA first-token -> x0
  mha_kernel<<<(N + 3) / 4, 64, 0, stream>>>(emb, in_w, in_b, out_w, out_b, x0, N);

  // 2) zero cnt + agg (contiguous region, multiple of 4 floats)
  fill_zero_kernel<<<4096, 256, 0, stream>>>(
      (float4*)cnt, ((long long)N * NRELS + (long long)N * NRELS * D_FEAT) / 4);

  // 3) layer-1 edge aggregation (also builds counts, reused by layer 2)
  edge_agg_kernel<<<(E + 3) / 4, 256, 0, stream>>>(x0, src, dst, etype, agg, cnt, E, 1);

  // 4) layer-1 einsum + root + bias + relu -> x1   (WMMA f32)
  rgcn_einsum_kernel<64, 0><<<(N + 127) / 128, 128, 0, stream>>>(
      agg, cnt, x0, bpack1, b1, x1, N);

  // 5) zero agg
  fill_zero_kernel<<<4096, 256, 0, stream>>>(
      (float4*)agg, (long long)N * NRELS * D_FEAT / 4);

  // 6) layer-2 edge aggregation
  edge_agg_kernel<<<(E + 3) / 4, 256, 0, stream>>>(x1, src, dst, etype, agg, cnt, E, 0);

  // 7) layer-2 einsum + root + bias + sigmoid -> d_out
  rgcn_einsum_kernel<16, 1><<<(N + 127) / 128, 128, 0, stream>>>(
      agg, cnt, x1, bpack2, b2, (float*)d_out, N);
}